// SurfaceCircleConv_16088947491408
// MI455X (gfx1250) — compile-verified
//
#include <hip/hip_runtime.h>
#include <math.h>

// ---- problem constants (match reference setup_inputs) ----
constexpr int B_  = 16;
constexpr int N_  = 8192;
constexpr int NP  = 2048;
constexpr int KN  = 32;      // neighbors per point
constexpr int D_  = 61;      // raw point feature dim
constexpr int C_  = 64;      // in_channel = D+3
constexpr int P_  = 5;       // partitions
constexpr int OC  = 128;     // out channel
constexpr int M_  = B_ * NP; // 32768 rows
constexpr int K1  = P_ * C_; // 320
constexpr float RADIUS_ = 1.5f;
constexpr float EPS_    = 1e-5f;

typedef __attribute__((ext_vector_type(2))) float v2f;
typedef __attribute__((ext_vector_type(8))) float v8f;

// ---------------------------------------------------------------------------
// 1) new_xyz = xyz gathered at data_idx  -> first 98304 floats of d_out
// ---------------------------------------------------------------------------
__global__ void new_xyz_kernel(const float* __restrict__ xyz,
                               const int* __restrict__ data_idx,
                               float* __restrict__ out) {
    int i = blockIdx.x * blockDim.x + threadIdx.x;      // over M_*3
    if (i >= M_ * 3) return;
    int j  = i % 3;
    int pt = i / 3;
    int b  = pt / NP;
    int n  = pt % NP;
    int src = data_idx[b * NP + n];
    out[i] = xyz[((size_t)b * N_ + src) * 3 + j];
}

// ---------------------------------------------------------------------------
// 2) gather neighbors + radial binning -> F[M_, 320]
//    one block (64 threads = channel id) per query point
// ---------------------------------------------------------------------------
__global__ void gather_bin_kernel(const float* __restrict__ xyz,
                                  const float* __restrict__ points,
                                  const float* __restrict__ lc,
                                  const int* __restrict__ nbr,
                                  float* __restrict__ F) {
    __shared__ int s_bin[KN];
    __shared__ int s_idx[KN];
    int pt = blockIdx.x;            // 0..M_-1
    int b  = pt / NP;
    int t  = threadIdx.x;           // 0..63 = channel
    if (t < KN) {
        const float* l = lc + ((size_t)pt * KN + t) * 3;
        float x = l[0], y = l[1];
        float d = sqrtf(x * x + y * y) * (1.0f / RADIUS_);
        d = fminf(d, 0.99f);
        s_bin[t] = (int)floorf(d * (float)P_);
        s_idx[t] = nbr[(size_t)pt * KN + t];
    }
    __syncthreads();
    float a0 = 0.f, a1 = 0.f, a2 = 0.f, a3 = 0.f, a4 = 0.f;
    int c = t;
    #pragma unroll 4
    for (int k = 0; k < KN; ++k) {
        int idx = s_idx[k];
        float v = (c < D_) ? points[((size_t)b * N_ + idx) * D_ + c]
                           : xyz[((size_t)b * N_ + idx) * 3 + (c - D_)];
        switch (s_bin[k]) {
            case 0: a0 += v; break;
            case 1: a1 += v; break;
            case 2: a2 += v; break;
            case 3: a3 += v; break;
            default: a4 += v; break;
        }
    }
    float* row = F + (size_t)pt * K1;
    row[0 * C_ + c] = a0;
    row[1 * C_ + c] = a1;
    row[2 * C_ + c] = a2;
    row[3 * C_ + c] = a3;
    row[4 * C_ + c] = a4;
}

// ---------------------------------------------------------------------------
// 3) fp32 WMMA GEMM:  Out[M_,128] = A[M_,Kd] @ Bm[Kd,128] + bias
//    block = 256 threads = 8 waves; wave w -> N-tile w (16 cols),
//    blockIdx.x -> M-tile (16 rows). V_WMMA_F32_16X16X4_F32, K step 4.
//    ISA layouts: A lanes 0-15 hold K=k,k+1; lanes 16-31 hold K=k+2,k+3.
//                 B same K split; N = lane&15.  C/D: M = r + 8*(lane>=16).
// ---------------------------------------------------------------------------
__global__ void wmma_gemm_f32_kernel(const float* __restrict__ A,
                                     const float* __restrict__ Bm,
                                     const float* __restrict__ bias,
                                     float* __restrict__ Out, int Kd) {
    int wave = threadIdx.x >> 5;        // 0..7
    int lane = threadIdx.x & 31;
    int half = lane >> 4;               // 0 or 1 (K sub-group)
    int l    = lane & 15;
    int m0   = blockIdx.x * 16;
    int n0   = wave * 16;

    v8f acc = {};
    const float* arow = A + (size_t)(m0 + l) * Kd;
    for (int k = 0; k < Kd; k += 4) {
        int ka = k + half * 2;
        v2f af, bf;
        af.x = arow[ka];
        af.y = arow[ka + 1];
        bf.x = Bm[(size_t)ka * OC + n0 + l];
        bf.y = Bm[(size_t)(ka + 1) * OC + n0 + l];
        acc = __builtin_amdgcn_wmma_f32_16x16x4_f32(
            /*neg_a=*/false, af, /*neg_b=*/false, bf,
            /*c_mod=*/(short)0, acc, /*reuse_a=*/false, /*reuse_b=*/false);
    }
    float bcol = bias[n0 + l];
    #pragma unroll
    for (int r = 0; r < 8; ++r) {
        int mrow = m0 + r + half * 8;
        Out[(size_t)mrow * OC + n0 + l] = acc[r] + bcol;
    }
}

// ---------------------------------------------------------------------------
// 4) deterministic two-pass column mean/var over M_ rows, 128 cols
// ---------------------------------------------------------------------------
__global__ void colreduce_pass1(const float* __restrict__ X,
                                float* __restrict__ Psum,
                                float* __restrict__ Pss) {
    int col = threadIdx.x;                       // 128 threads
    int rows_per_blk = M_ / 256;                 // 128
    int r0 = blockIdx.x * rows_per_blk;
    float s = 0.f, ss = 0.f;
    for (int r = 0; r < rows_per_blk; ++r) {
        float v = X[(size_t)(r0 + r) * OC + col];
        s += v;
        ss += v * v;
    }
    Psum[blockIdx.x * OC + col] = s;
    Pss[blockIdx.x * OC + col]  = ss;
}

__global__ void colreduce_pass2(const float* __restrict__ Psum,
                                const float* __restrict__ Pss,
                                float* __restrict__ stats) {
    int col = threadIdx.x;                       // 128 threads, 1 block
    float s = 0.f, ss = 0.f;
    for (int i = 0; i < 256; ++i) {
        s += Psum[i * OC + col];
        ss += Pss[i * OC + col];
    }
    float mu  = s / (float)M_;
    float var = ss / (float)M_ - mu * mu;        // population var (ddof=0)
    stats[col]      = mu;
    stats[OC + col] = var;
}

// ---------------------------------------------------------------------------
// 5) fused batch-norm (batch stats) + ReLU, elementwise
// ---------------------------------------------------------------------------
__global__ void bn_relu_kernel(const float* __restrict__ X,
                               const float* __restrict__ stats,
                               const float* __restrict__ g,
                               const float* __restrict__ be,
                               float* __restrict__ Outp) {
    int i = blockIdx.x * blockDim.x + threadIdx.x;
    if (i >= M_ * OC) return;
    int col = i & (OC - 1);
    float mu  = stats[col];
    float var = stats[OC + col];
    float v = g[col] * (X[i] - mu) * rsqrtf(var + EPS_) + be[col];
    Outp[i] = fmaxf(v, 0.0f);
}

// ---------------------------------------------------------------------------
extern "C" void kernel_launch(void* const* d_in, const int* in_sizes, int n_in,
                              void* d_out, int out_size, void* d_ws, size_t ws_size,
                              hipStream_t stream) {
    (void)in_sizes; (void)n_in; (void)out_size; (void)ws_size;
    const float* xyz  = (const float*)d_in[0];
    const float* pts  = (const float*)d_in[1];
    const float* lc   = (const float*)d_in[2];
    const int*   nbr  = (const int*)d_in[3];
    // d_in[4] = parameter_list (unused by reference math)
    const int*   didx = (const int*)d_in[5];
    const float* W1   = (const float*)d_in[6];
    const float* b1   = (const float*)d_in[7];
    const float* g1   = (const float*)d_in[8];
    const float* be1  = (const float*)d_in[9];
    const float* W2   = (const float*)d_in[10];
    const float* b2   = (const float*)d_in[11];
    const float* g2   = (const float*)d_in[12];
    const float* be2  = (const float*)d_in[13];

    float* out_xyz = (float*)d_out;                 // [M_,3]
    float* out_pts = (float*)d_out + (size_t)M_ * 3; // [M_,128]

    // workspace layout (bytes)
    char* ws = (char*)d_ws;
    float* F    = (float*)(ws);                                   // M_*320
    float* X1   = (float*)(ws + (size_t)M_ * K1 * 4);             // M_*128
    float* Xn   = (float*)(ws);                                   // reuse F region
    float* Y    = (float*)(ws + (size_t)M_ * OC * 4);             // reuse F region (16MB in)
    char*  tail = ws + (size_t)M_ * K1 * 4 + (size_t)M_ * OC * 4; // after X1
    float* Psum   = (float*)(tail);                               // 256*128
    float* Pss    = (float*)(tail + 256 * OC * 4);                // 256*128
    float* stats1 = (float*)(tail + 2 * 256 * OC * 4);            // 256 floats
    float* stats2 = stats1 + 2 * OC;

    // 1) new_xyz
    new_xyz_kernel<<<(M_ * 3 + 255) / 256, 256, 0, stream>>>(xyz, didx, out_xyz);

    // 2) gather + radial bin -> F
    gather_bin_kernel<<<M_, 64, 0, stream>>>(xyz, pts, lc, nbr, F);

    // 3) GEMM1 (WMMA f32): X1 = F @ W1 + b1
    wmma_gemm_f32_kernel<<<M_ / 16, 256, 0, stream>>>(F, W1, b1, X1, K1);

    // 4) BN1 stats + normalize+ReLU -> Xn
    colreduce_pass1<<<256, OC, 0, stream>>>(X1, Psum, Pss);
    colreduce_pass2<<<1, OC, 0, stream>>>(Psum, Pss, stats1);
    bn_relu_kernel<<<(M_ * OC) / 256, 256, 0, stream>>>(X1, stats1, g1, be1, Xn);

    // 5) GEMM2 (WMMA f32): Y = Xn @ W2 + b2
    wmma_gemm_f32_kernel<<<M_ / 16, 256, 0, stream>>>(Xn, W2, b2, Y, OC);

    // 6) BN2 stats + normalize+ReLU -> d_out
    colreduce_pass1<<<256, OC, 0, stream>>>(Y, Psum, Pss);
    colreduce_pass2<<<1, OC, 0, stream>>>(Psum, Pss, stats2);
    bn_relu_kernel<<<(M_ * OC) / 256, 256, 0, stream>>>(Y, stats2, g2, be2, out_pts);
}